// Model_6176162971964
// MI455X (gfx1250) — compile-verified
//
#include <hip/hip_runtime.h>
#include <hip/hip_bf16.h>

typedef __attribute__((ext_vector_type(2))) float v2f;
typedef __attribute__((ext_vector_type(8))) float v8f;

#define B_  32
#define S_  64
#define D_  256
#define H_  8
#define HD_ 32
#define NS_ 65

// ---------------- wave32 reductions ----------------
__device__ inline float wave_sum(float v) {
#pragma unroll
  for (int off = 16; off > 0; off >>= 1) v += __shfl_down(v, off, 32);
  return v;
}
__device__ inline float wave_max(float v) {
#pragma unroll
  for (int off = 16; off > 0; off >>= 1) v = fmaxf(v, __shfl_down(v, off, 32));
  return v;
}

// ---------------- fp32 WMMA GEMM (batched): C[z][M,N] = A[z][M,K] @ W[z][N,K]^T + bias ----
// One 16x16 C tile per wave, 8 waves per block (block = 128 rows x 16 cols).
// The 16-column W tile is shared by all 8 waves -> staged once in LDS (K must be 256).
__global__ void gemm_wmma_f32(const float* __restrict__ A, int lda, size_t strideA,
                              const float* __restrict__ W, int ldw, size_t strideW,
                              const float* __restrict__ bias,
                              float* __restrict__ C, size_t strideC,
                              int M, int N, int K) {
  A += (size_t)blockIdx.z * strideA;
  W += (size_t)blockIdx.z * strideW;
  C += (size_t)blockIdx.z * strideC;

  const int lane = threadIdx.x & 31;
  const int wid  = threadIdx.x >> 5;
  const int n0 = blockIdx.x * 16;
  const int m0 = (blockIdx.y * 8 + wid) * 16;

  __shared__ float Wl[16 * 256];          // [col][k], K == 256
  for (int idx = threadIdx.x; idx < 16 * K; idx += 256)
    Wl[idx] = W[(n0 + (idx >> 8)) * ldw + (idx & 255)];
  __syncthreads();

  if (m0 < M) {                           // wave-uniform: EXEC all-ones around WMMA
    const int row   = m0 + (lane & 15);   // A lane row (M)
    const int colk  = (lane & 15) << 8;   // LDS base of this lane's column
    const int khalf = (lane >> 4) * 2;    // lanes 0-15 -> K 0,1 ; lanes 16-31 -> K 2,3
    __builtin_prefetch(A + (size_t)row * lda, 0, 1);   // global_prefetch_b8

    v8f acc = {};
    for (int k0 = 0; k0 < K; k0 += 4) {
      v2f a, b;
      a.x = A[(size_t)row * lda + k0 + khalf];
      a.y = A[(size_t)row * lda + k0 + khalf + 1];
      b.x = Wl[colk + k0 + khalf];
      b.y = Wl[colk + k0 + khalf + 1];
      acc = __builtin_amdgcn_wmma_f32_16x16x4_f32(false, a, false, b,
                                                  (short)0, acc, false, false);
    }
    const int col = n0 + (lane & 15);
    const float bv = bias ? bias[col] : 0.0f;
    const int rbase = m0 + (lane >> 4) * 8;
#pragma unroll
    for (int r = 0; r < 8; ++r)
      C[(size_t)(rbase + r) * N + col] = acc[r] + bv;
  }
}

// ---------------- per-row inverse norms of desc and var ----------------
__global__ void norms_kernel(const float* __restrict__ desc,
                             const float* __restrict__ nve,
                             float* __restrict__ invd, float* __restrict__ invv) {
  const int bs = blockIdx.x;                // (b*S + s)
  const int b  = bs / S_, s = bs % S_;
  const int t  = threadIdx.x;
  const int lane = t & 31, wid = t >> 5;
  const float x = desc[bs * D_ + t];
  const float y = nve[(b * NS_ + 1 + s) * D_ + t];
  float s1 = wave_sum(x * x);
  float s2 = wave_sum(y * y);
  __shared__ float red[16];
  if (lane == 0) { red[wid] = s1; red[8 + wid] = s2; }
  __syncthreads();
  if (t == 0) {
    float a1 = 0.f, a2 = 0.f;
#pragma unroll
    for (int w = 0; w < 8; ++w) { a1 += red[w]; a2 += red[8 + w]; }
    invd[bs] = rsqrtf(a1);
    invv[bs] = rsqrtf(a2);
  }
}

// ---------------- per-row sums / sums-of-squares of u and w' ----------------
__global__ void rowstats_kernel(const float* __restrict__ u, const float* __restrict__ wp,
                                float* __restrict__ su, float* __restrict__ squ,
                                float* __restrict__ sw, float* __restrict__ sqw) {
  const int bs = blockIdx.x;
  const int t = threadIdx.x;
  const int lane = t & 31, wid = t >> 5;
  const float uv = u[bs * D_ + t];
  const float wv = wp[bs * D_ + t];
  float s1 = wave_sum(uv);
  float s2 = wave_sum(uv * uv);
  float s3 = wave_sum(wv);
  float s4 = wave_sum(wv * wv);
  __shared__ float red[32];
  if (lane == 0) { red[wid] = s1; red[8 + wid] = s2; red[16 + wid] = s3; red[24 + wid] = s4; }
  __syncthreads();
  if (t == 0) {
    float a1 = 0.f, a2 = 0.f, a3 = 0.f, a4 = 0.f;
#pragma unroll
    for (int w = 0; w < 8; ++w) {
      a1 += red[w]; a2 += red[8 + w]; a3 += red[16 + w]; a4 += red[24 + w];
    }
    su[bs] = a1; squ[bs] = a2; sw[bs] = a3; sqw[bs] = a4;
  }
}

// ---------------- adjacency = softmax_j( (gsim+tb>0 && i!=j) ? ssim : 0 ) ----------------
__global__ void adj_kernel(const float* __restrict__ desc,
                           const float* __restrict__ nve,
                           const float* __restrict__ invd,
                           const float* __restrict__ invv,
                           const float* __restrict__ tb,
                           float* __restrict__ adj) {
  const int bi = blockIdx.x;                // (b*S + i)
  const int b = bi / S_, i = bi % S_;
  const int t = threadIdx.x;
  const int j = t >> 2, part = t & 3;
  const float tb0 = tb[0];

  const float* di = desc + (b * S_ + i) * D_;
  const float* dj = desc + (b * S_ + j) * D_;
  const float* vi = nve + (b * NS_ + 1 + i) * D_;
  const float* vj = nve + (b * NS_ + 1 + j) * D_;
  float gd = 0.f, sd = 0.f;
  for (int c = part * 64; c < part * 64 + 64; ++c) {
    gd += di[c] * dj[c];
    sd += vi[c] * vj[c];
  }
  __shared__ float gred[256], sred[256], vals[64];
  gred[t] = gd; sred[t] = sd;
  __syncthreads();
  if (part == 0) {
    const float g  = (gred[t] + gred[t + 1] + gred[t + 2] + gred[t + 3])
                       * invd[b * S_ + i] * invd[b * S_ + j] + tb0;
    const float ss = (sred[t] + sred[t + 1] + sred[t + 2] + sred[t + 3])
                       * invv[b * S_ + i] * invv[b * S_ + j];
    vals[j] = (g > 0.f && j != i) ? ss : 0.f;   // A*ssim, A binary straight-through
  }
  __syncthreads();
  if (t < 32) {
    const float v0 = vals[t], v1 = vals[t + 32];
    float mx = wave_max(fmaxf(v0, v1));
    mx = __shfl(mx, 0, 32);
    const float e0 = expf(v0 - mx), e1 = expf(v1 - mx);
    float s = wave_sum(e0 + e1);
    s = __shfl(s, 0, 32);
    const float inv = 1.f / s;
    adj[(b * S_ + i) * S_ + t]      = e0 * inv;
    adj[(b * S_ + i) * S_ + t + 32] = e1 * inv;
  }
}

// ---------------- sq[b,h,i] = q . wq  ;  sk[b,h,i] = k . wk ----------------
__global__ void qk_kernel(const float* __restrict__ qb, const float* __restrict__ kb,
                          const float* __restrict__ Wa,
                          float* __restrict__ sqb, float* __restrict__ skb) {
  const int bi = blockIdx.x;                // (b*NS + i)
  const int b = bi / NS_, i = bi % NS_;
  const int h = threadIdx.x >> 5, d = threadIdx.x & 31;
  float qv = qb[bi * D_ + h * HD_ + d] * Wa[d];
  float kv = kb[bi * D_ + h * HD_ + d] * Wa[HD_ + d];
  qv = wave_sum(qv);
  kv = wave_sum(kv);
  if (d == 0) {
    sqb[(b * H_ + h) * NS_ + i] = qv;
    skb[(b * H_ + h) * NS_ + i] = kv;
  }
}

// ---------------- fold We2/be2 with we: M[h,c], bconst[h] ----------------
__global__ void m8_kernel(const float* __restrict__ We2, const float* __restrict__ be2,
                          const float* __restrict__ Wa,
                          float* __restrict__ M8, float* __restrict__ bconst) {
  const int c = threadIdx.x;
  for (int h = 0; h < H_; ++h) {
    float acc = 0.f;
    for (int d = 0; d < HD_; ++d) acc += We2[(h * HD_ + d) * D_ + c] * Wa[2 * HD_ + d];
    M8[h * D_ + c] = acc;
  }
  if (c < H_) {
    float acc = 0.f;
    for (int d = 0; d < HD_; ++d) acc += be2[c * HD_ + d] * Wa[2 * HD_ + d];
    bconst[c] = acc;
  }
}

// ---------------- fused edge-scores + softmax + ctx + output projection ----------------
// One block per (b, i); 256 threads (thread t == channel c; wave wid == head).
// LN statistics come from precomputed row sums + the WMMA-computed cross matrix:
//   var(u_i + w'_j) uses  sum(u)+sum(w'),  sum(u^2)+sum(w'^2)+2*(u_i . w'_j).
__global__ void fused_attn(const float* __restrict__ u, const float* __restrict__ wp,
                           const float* __restrict__ adj,
                           const float* __restrict__ su, const float* __restrict__ squ,
                           const float* __restrict__ sw, const float* __restrict__ sqw,
                           const float* __restrict__ cross,
                           const float* __restrict__ sqb, const float* __restrict__ skb,
                           const float* __restrict__ M8, const float* __restrict__ bconst,
                           const float* __restrict__ vbuf,
                           const float* __restrict__ lng, const float* __restrict__ lnb,
                           const float* __restrict__ ba,
                           const float* __restrict__ Wo, const float* __restrict__ bo,
                           float* __restrict__ outbuf, float* __restrict__ awbuf) {
  const int b = blockIdx.x / NS_;
  const int i = blockIdx.x % NS_;
  const int t = threadIdx.x;
  const int lane = t & 31, wid = t >> 5;

  __shared__ float M8_s[H_ * D_];
  __shared__ float bc_s[H_];
  __shared__ float sq_s[H_];
  __shared__ float sk_s[H_ * NS_];
  __shared__ float sc_s[H_ * NS_];
  __shared__ float rl_s[D_];
  __shared__ float ctx_s[D_];
  __shared__ float su_s[S_], squ_s[S_], sw_s[S_], sqw_s[S_], cr_s[S_], adj_s[S_];

  for (int idx = t; idx < H_ * D_; idx += 256) M8_s[idx] = M8[idx];
  if (t < H_) {
    bc_s[t] = bconst[t];
    sq_s[t] = sqb[(b * H_ + t) * NS_ + i];
  }
  for (int idx = t; idx < H_ * NS_; idx += 256)
    sk_s[idx] = skb[(b * H_ + idx / NS_) * NS_ + (idx % NS_)];
  if (t < S_) {
    su_s[t]  = su[b * S_ + t];
    squ_s[t] = squ[b * S_ + t];
    sw_s[t]  = sw[b * S_ + t];
    sqw_s[t] = sqw[b * S_ + t];
    if (i == 0) {
      cr_s[t]  = cross[((size_t)(b * S_ + t)) * S_ + t];   // diagonal u_j . w'_j
      adj_s[t] = 1.f;                                      // na[:,0,1:] == 1
    } else {
      cr_s[t]  = cross[((size_t)(b * S_ + (i - 1))) * S_ + t];
      adj_s[t] = adj[(b * S_ + (i - 1)) * S_ + t];
    }
  }

  const float ba0  = ba[0];
  const float lngc = lng[t], lnbc = lnb[t];
  const float ac_i = (i >= 1) ? u[(b * S_ + (i - 1)) * D_ + t] : 0.f;
  __syncthreads();

  // j == 0 column: na == 0 -> se == 0, mask -1e9
  if (t < H_) sc_s[t * NS_ + 0] = sq_s[t] + sk_s[t * NS_ + 0] + ba0 - 1.0e9f;

  for (int j = 1; j < NS_; ++j) {
    const int jj = j - 1;
    const int ri = (i == 0) ? jj : (i - 1);
    const float ac = (i == 0) ? u[(b * S_ + jj) * D_ + t] : ac_i;
    const float r  = ac + wp[(b * S_ + jj) * D_ + t];

    const float sum = su_s[ri] + sw_s[jj];
    const float ssq = squ_s[ri] + sqw_s[jj] + 2.f * cr_s[jj];
    const float mu  = sum * (1.f / D_);
    const float var = ssq * (1.f / D_) - mu * mu;
    const float inv = rsqrtf(var + 1e-5f);
    rl_s[t] = fmaxf((r - mu) * inv * lngc + lnbc, 0.f);
    __syncthreads();

    // per-head dot: wave 'wid' computes relu-row . M[wid,:]
    float p = 0.f;
#pragma unroll
    for (int q8 = 0; q8 < 8; ++q8)
      p += rl_s[lane + 32 * q8] * M8_s[wid * D_ + lane + 32 * q8];
    p = wave_sum(p);
    if (lane == 0)
      sc_s[wid * NS_ + j] = sq_s[wid] + sk_s[wid * NS_ + j]
                          + adj_s[jj] * (p + bc_s[wid]) + ba0;
    __syncthreads();   // protect rl_s for next iteration
  }

  // per-head softmax over j (wave 'wid' owns head 'wid')
  {
    float mx = -3.0e38f;
    for (int j = lane; j < NS_; j += 32) mx = fmaxf(mx, sc_s[wid * NS_ + j]);
    mx = wave_max(mx);
    mx = __shfl(mx, 0, 32);
    float s = 0.f;
    for (int j = lane; j < NS_; j += 32) {
      const float e = expf(sc_s[wid * NS_ + j] - mx);
      sc_s[wid * NS_ + j] = e;
      s += e;
    }
    s = wave_sum(s);
    s = __shfl(s, 0, 32);
    const float invs = 1.f / s;
    for (int j = lane; j < NS_; j += 32) {
      const float aw = sc_s[wid * NS_ + j] * invs;
      sc_s[wid * NS_ + j] = aw;
      awbuf[((size_t)(b * H_ + wid) * NS_ + i) * NS_ + j] = aw;
    }
  }
  __syncthreads();

  // ctx[h, d] = sum_j aw[h,j] * v[b,h,j,d]
  {
    float acc = 0.f;
    for (int j = 0; j < NS_; ++j)
      acc += sc_s[wid * NS_ + j] * vbuf[(b * NS_ + j) * D_ + wid * HD_ + lane];
    ctx_s[t] = acc;
  }
  __syncthreads();

  // out[b,i,o] = ctx . Wo[o,:] + bo[o]
  {
    float val = bo[t];
    for (int c = 0; c < D_; ++c) val += ctx_s[c] * Wo[t * D_ + c];
    outbuf[(b * NS_ + i) * D_ + t] = val;
  }
}

// ---------------- launcher ----------------
extern "C" void kernel_launch(void* const* d_in, const int* in_sizes, int n_in,
                              void* d_out, int out_size, void* d_ws, size_t ws_size,
                              hipStream_t stream) {
  const float* desc_emb = (const float*)d_in[0];
  const float* nve      = (const float*)d_in[1];
  const float* Wg  = (const float*)d_in[2];
  const float* bg  = (const float*)d_in[3];
  const float* Wq  = (const float*)d_in[4];
  const float* bq  = (const float*)d_in[5];
  const float* Wk  = (const float*)d_in[6];
  const float* bk  = (const float*)d_in[7];
  const float* Wv  = (const float*)d_in[8];
  const float* bv  = (const float*)d_in[9];
  const float* We1 = (const float*)d_in[10];
  const float* be1 = (const float*)d_in[11];
  const float* lng = (const float*)d_in[12];
  const float* lnb = (const float*)d_in[13];
  const float* We2 = (const float*)d_in[14];
  const float* be2 = (const float*)d_in[15];
  const float* Wa  = (const float*)d_in[16];
  const float* ba  = (const float*)d_in[17];
  const float* Wo  = (const float*)d_in[18];
  const float* bo  = (const float*)d_in[19];
  const float* tb  = (const float*)d_in[20];

  float* ws = (float*)d_ws;
  float* descb = ws;                 ws += (size_t)B_ * S_ * D_;   // 524288
  float* qbuf  = ws;                 ws += (size_t)B_ * NS_ * D_;  // 532480
  float* kbuf  = ws;                 ws += (size_t)B_ * NS_ * D_;
  float* vbuf  = ws;                 ws += (size_t)B_ * NS_ * D_;
  float* ubuf  = ws;                 ws += (size_t)B_ * S_ * D_;
  float* wpbuf = ws;                 ws += (size_t)B_ * S_ * D_;
  float* invd  = ws;                 ws += (size_t)B_ * S_;
  float* invv  = ws;                 ws += (size_t)B_ * S_;
  float* adjb  = ws;                 ws += (size_t)B_ * S_ * S_;
  float* sqb   = ws;                 ws += (size_t)B_ * H_ * NS_;
  float* skb   = ws;                 ws += (size_t)B_ * H_ * NS_;
  float* M8b   = ws;                 ws += (size_t)H_ * D_;
  float* bcb   = ws;                 ws += (size_t)H_;
  float* sub   = ws;                 ws += (size_t)B_ * S_;
  float* squb  = ws;                 ws += (size_t)B_ * S_;
  float* swb   = ws;                 ws += (size_t)B_ * S_;
  float* sqwb  = ws;                 ws += (size_t)B_ * S_;
  float* crossb= ws;                 ws += (size_t)B_ * S_ * S_;

  float* outp = (float*)d_out;                       // (B, NS, D)
  float* awp  = outp + (size_t)B_ * NS_ * D_;        // (B, H, NS, NS)

  const dim3 blk(256);
  const int MD = B_ * S_;     // 2048 rows
  const int MN = B_ * NS_;    // 2080 rows

  // desc = desc_emb @ Wg^T + bg
  gemm_wmma_f32<<<dim3(D_ / 16, (MD + 127) / 128, 1), blk, 0, stream>>>(
      desc_emb, D_, 0, Wg, D_, 0, bg, descb, 0, MD, D_, D_);
  // q,k,v projections of nve
  gemm_wmma_f32<<<dim3(D_ / 16, (MN + 127) / 128, 1), blk, 0, stream>>>(
      nve, D_, 0, Wq, D_, 0, bq, qbuf, 0, MN, D_, D_);
  gemm_wmma_f32<<<dim3(D_ / 16, (MN + 127) / 128, 1), blk, 0, stream>>>(
      nve, D_, 0, Wk, D_, 0, bk, kbuf, 0, MN, D_, D_);
  gemm_wmma_f32<<<dim3(D_ / 16, (MN + 127) / 128, 1), blk, 0, stream>>>(
      nve, D_, 0, Wv, D_, 0, bv, vbuf, 0, MN, D_, D_);
  // u = desc @ We1[:, :D]^T ; w' = desc @ We1[:, D:]^T + be1   (We1 row stride 2D)
  gemm_wmma_f32<<<dim3(D_ / 16, (MD + 127) / 128, 1), blk, 0, stream>>>(
      descb, D_, 0, We1, 2 * D_, 0, nullptr, ubuf, 0, MD, D_, D_);
  gemm_wmma_f32<<<dim3(D_ / 16, (MD + 127) / 128, 1), blk, 0, stream>>>(
      descb, D_, 0, We1 + D_, 2 * D_, 0, be1, wpbuf, 0, MD, D_, D_);
  // cross[b] = U[b] @ W'[b]^T  (batched, per-batch 64x64x256) -> LN variance cross terms
  gemm_wmma_f32<<<dim3(S_ / 16, 1, B_), blk, 0, stream>>>(
      ubuf, D_, (size_t)S_ * D_, wpbuf, D_, (size_t)S_ * D_, nullptr,
      crossb, (size_t)S_ * S_, S_, S_, D_);

  norms_kernel<<<B_ * S_, blk, 0, stream>>>(descb, nve, invd, invv);
  rowstats_kernel<<<B_ * S_, blk, 0, stream>>>(ubuf, wpbuf, sub, squb, swb, sqwb);
  adj_kernel<<<B_ * S_, blk, 0, stream>>>(descb, nve, invd, invv, tb, adjb);
  qk_kernel<<<B_ * NS_, blk, 0, stream>>>(qbuf, kbuf, Wa, sqb, skb);
  m8_kernel<<<1, blk, 0, stream>>>(We2, be2, Wa, M8b, bcb);

  fused_attn<<<B_ * NS_, blk, 0, stream>>>(ubuf, wpbuf, adjb,
                                           sub, squb, swb, sqwb, crossb,
                                           sqb, skb, M8b, bcb,
                                           vbuf, lng, lnb, ba, Wo, bo, outp, awp);
}